// EMDLoss_62981400428805
// MI455X (gfx1250) — compile-verified
//
#include <hip/hip_runtime.h>
#include <math.h>

// ---------------------------------------------------------------------------
// Sinkhorn-EMD (SimCC) for MI455X / gfx1250.
//   - wave32: one wave per (b,k) problem; 768 bins = 24 f32 per lane.
//   - gfx1250 async copy path: global_load_async_to_lds_b128 double-buffers
//     the next problem's 3KB pred tile behind the 10 Sinkhorn iterations,
//     synchronized with s_wait_asynccnt (ASYNCcnt).
//   - No WMMA: the math has no matmul structure (768-point logsumexp
//     reductions per problem); this kernel is transcendental-bound.
// ---------------------------------------------------------------------------

#define SK_EPS     0.1f
#define SK_INVEPS  10.0f
#define SK_ITERS   10
#define SK_N       768
#define SK_CHUNKS  6          // 768 = 6 * 128 (128 floats per wave32 float4 round)
#define WPB        8          // waves per block (256 threads, wave32)

__device__ __forceinline__ float wave_sum32(float x) {
#pragma unroll
  for (int off = 16; off > 0; off >>= 1)
    x += __shfl_xor(x, off, 32);
  return x;
}

__device__ __forceinline__ float wave_max32(float x) {
#pragma unroll
  for (int off = 16; off > 0; off >>= 1)
    x = fmaxf(x, __shfl_xor(x, off, 32));
  return x;
}

// Stage one problem's 768 floats into LDS with 6 async b128 copies per lane.
// lds_base is the wave-relative LDS byte address for this lane (16B aligned).
__device__ __forceinline__ void async_stage_tile(unsigned lds_base,
                                                 const float* g) {
#pragma unroll
  for (int ch = 0; ch < SK_CHUNKS; ++ch) {
    asm volatile("global_load_async_to_lds_b128 %0, %1, off"
                 :: "v"(lds_base + (unsigned)(ch * 512)),
                    "v"(g + ch * 128)
                 : "memory");
  }
}

__global__ void __launch_bounds__(WPB * 32)
sinkhorn_emd_kernel(const float* __restrict__ preds,
                    const float* __restrict__ targets,
                    float* __restrict__ per_prob,
                    int n_prob) {
  __shared__ __align__(16) float smem[WPB][2][SK_N];

  const int lane = threadIdx.x & 31;
  const int wv   = threadIdx.x >> 5;
  const int wid  = blockIdx.x * WPB + wv;
  const int wstr = gridDim.x * WPB;

  // Low 32 bits of a generic shared pointer == wave-relative LDS byte address.
  const unsigned lb0 =
      (unsigned)(unsigned long long)(&smem[wv][0][0]) + (unsigned)(lane * 16);
  const unsigned lb1 =
      (unsigned)(unsigned long long)(&smem[wv][1][0]) + (unsigned)(lane * 16);

  int p = wid;
  if (p < n_prob)
    async_stage_tile(lb0, preds + (size_t)p * SK_N + lane * 4);

  int buf = 0;
  for (; p < n_prob; p += wstr, buf ^= 1) {
    // Wait for the tile staged for this problem (per-wave ASYNCcnt).
    asm volatile("s_wait_asynccnt 0" ::: "memory");

    // Pull 24 values/lane out of LDS: n = ch*128 + lane*4 + c.
    float rp[SK_CHUNKS][4];
#pragma unroll
    for (int ch = 0; ch < SK_CHUNKS; ++ch) {
      const float4 q = *reinterpret_cast<const float4*>(
          &smem[wv][buf][ch * 128 + lane * 4]);
      rp[ch][0] = q.x; rp[ch][1] = q.y; rp[ch][2] = q.z; rp[ch][3] = q.w;
    }

    // Kick off the prefetch for the next problem; it completes under the
    // ~10 Sinkhorn iterations of VALU/transcendental work below.
    const int pn = p + wstr;
    if (pn < n_prob)
      async_stage_tile(buf ? lb0 : lb1, preds + (size_t)pn * SK_N + lane * 4);

    // Target-side marginal: w_y = [d2-t, t-d1] at support {d1, d2}.
    const float t   = targets[p];
    const float d1  = floorf(t);
    const float d2  = fminf(d1 + 1.0f, (float)(SK_N - 1));
    const float lw0 = __logf(d2 - t);
    const float lw1 = __logf(t - d1);

    // log w_x = log(relu(p)) - log(sum relu(p))   (log(0) -> -inf, as in JAX)
    float part = 0.0f;
#pragma unroll
    for (int ch = 0; ch < SK_CHUNKS; ++ch)
#pragma unroll
      for (int c = 0; c < 4; ++c) {
        const float r = fmaxf(rp[ch][c], 0.0f);
        rp[ch][c] = r;
        part += r;
      }
    const float logtot = __logf(wave_sum32(part));
#pragma unroll
    for (int ch = 0; ch < SK_CHUNKS; ++ch)
#pragma unroll
      for (int c = 0; c < 4; ++c)
        rp[ch][c] = __logf(rp[ch][c]) - logtot;

    const float xb = (float)(lane * 4);
    float u[SK_CHUNKS][4];
    float v0 = 0.0f, v1 = 0.0f;

    for (int it = 0; it < SK_ITERS; ++it) {
      // u_n = eps*(log wx_n - lse_j((v_j - |n-y_j|)/eps)); 2-way stable lse.
#pragma unroll
      for (int ch = 0; ch < SK_CHUNKS; ++ch)
#pragma unroll
        for (int c = 0; c < 4; ++c) {
          const float xn = xb + (float)(ch * 128 + c);
          const float a0 = (v0 - fabsf(xn - d1)) * SK_INVEPS;
          const float a1 = (v1 - fabsf(xn - d2)) * SK_INVEPS;
          const float m  = fmaxf(a0, a1);
          const float lse = m + __logf(1.0f + __expf(-fabsf(a0 - a1)));
          u[ch][c] = SK_EPS * rp[ch][c] - SK_EPS * lse;
        }
      // v_j = eps*(log wy_j - lse_n((u_n - |n-y_j|)/eps)); max/sum over 768.
      float m0 = -INFINITY, m1 = -INFINITY;
#pragma unroll
      for (int ch = 0; ch < SK_CHUNKS; ++ch)
#pragma unroll
        for (int c = 0; c < 4; ++c) {
          const float xn = xb + (float)(ch * 128 + c);
          m0 = fmaxf(m0, (u[ch][c] - fabsf(xn - d1)) * SK_INVEPS);
          m1 = fmaxf(m1, (u[ch][c] - fabsf(xn - d2)) * SK_INVEPS);
        }
      m0 = wave_max32(m0);
      m1 = wave_max32(m1);
      float s0 = 0.0f, s1 = 0.0f;
#pragma unroll
      for (int ch = 0; ch < SK_CHUNKS; ++ch)
#pragma unroll
        for (int c = 0; c < 4; ++c) {
          const float xn = xb + (float)(ch * 128 + c);
          s0 += __expf((u[ch][c] - fabsf(xn - d1)) * SK_INVEPS - m0);
          s1 += __expf((u[ch][c] - fabsf(xn - d2)) * SK_INVEPS - m1);
        }
      s0 = wave_sum32(s0);
      s1 = wave_sum32(s1);
      v0 = SK_EPS * lw0 - SK_EPS * (m0 + __logf(s0));
      v1 = SK_EPS * lw1 - SK_EPS * (m1 + __logf(s1));
    }

    // loss_p = sum_{n,j} exp((u_n + v_j - C_nj)/eps) * C_nj
    float acc = 0.0f;
#pragma unroll
    for (int ch = 0; ch < SK_CHUNKS; ++ch)
#pragma unroll
      for (int c = 0; c < 4; ++c) {
        const float xn = xb + (float)(ch * 128 + c);
        const float c0 = fabsf(xn - d1);
        const float c1 = fabsf(xn - d2);
        acc += __expf((u[ch][c] + v0 - c0) * SK_INVEPS) * c0;
        acc += __expf((u[ch][c] + v1 - c1) * SK_INVEPS) * c1;
      }
    acc = wave_sum32(acc);
    if (lane == 0) per_prob[p] = acc;
  }
}

// Deterministic single-block reduction of per-problem losses (no float
// atomics: the harness requires identical output on every replay).
__global__ void __launch_bounds__(256)
final_reduce_kernel(const float* __restrict__ per_prob,
                    float* __restrict__ out, int n) {
  __shared__ float sb[256];
  float a = 0.0f;
  for (int i = threadIdx.x; i < n; i += 256) a += per_prob[i];
  sb[threadIdx.x] = a;
  __syncthreads();
#pragma unroll
  for (int s = 128; s > 0; s >>= 1) {
    if ((int)threadIdx.x < s) sb[threadIdx.x] += sb[threadIdx.x + s];
    __syncthreads();
  }
  if (threadIdx.x == 0) out[0] = sb[0];
}

extern "C" void kernel_launch(void* const* d_in, const int* in_sizes, int n_in,
                              void* d_out, int out_size, void* d_ws, size_t ws_size,
                              hipStream_t stream) {
  (void)n_in; (void)out_size; (void)ws_size;
  const float* preds   = (const float*)d_in[0];   // (B,K,768) f32
  const float* targets = (const float*)d_in[1];   // (B,K,1)   f32
  const int n_prob = in_sizes[1];                 // B*K = 4352
  float* per_prob = (float*)d_ws;                 // 4352 f32 scratch

  // ~2 problems per wave so the double-buffered async prefetch overlaps
  // the Sinkhorn compute of the current problem.
  int waves  = (n_prob + 1) / 2;
  int blocks = (waves + WPB - 1) / WPB;
  if (blocks < 1) blocks = 1;

  sinkhorn_emd_kernel<<<blocks, WPB * 32, 0, stream>>>(preds, targets,
                                                       per_prob, n_prob);
  final_reduce_kernel<<<1, 256, 0, stream>>>(per_prob, (float*)d_out, n_prob);
}